// U3Layer_54589034332666
// MI455X (gfx1250) — compile-verified
//
#include <hip/hip_runtime.h>

typedef __attribute__((ext_vector_type(2))) float v2f;
typedef __attribute__((ext_vector_type(8))) float v8f;

// d_ws float layout: Ar[4096] | Ai[4096] | Br[4096] | Bi[4096]   (64 KB)
// A = kron(g0..g5) acts on high 6 qubits (row index i), B = kron(g6..g11) on low 6 (j).

__global__ __launch_bounds__(256) void u3_build_ops(const float* __restrict__ thetas,
                                                    float* __restrict__ ws) {
  int t = blockIdx.x * blockDim.x + threadIdx.x;
  if (t >= 2 * 4096) return;
  int mat = t >> 12;        // 0 = A, 1 = B
  int idx = t & 4095;
  int i = idx >> 6;         // row (0..63)
  int j = idx & 63;         // col (0..63)
  int qbase = mat * 6;
  float re = 1.0f, im = 0.0f;
#pragma unroll
  for (int q = 0; q < 6; ++q) {
    int bi = (i >> (5 - q)) & 1;
    int bj = (j >> (5 - q)) & 1;
    float th = thetas[(qbase + q) * 3 + 0];
    float ph = thetas[(qbase + q) * 3 + 1];
    float la = thetas[(qbase + q) * 3 + 2];
    float s, c;
    __sincosf(th * 0.5f, &s, &c);
    float gr, gi;
    if (!bi && !bj) {
      gr = c; gi = 0.0f;                       // cos(t/2)
    } else if (!bi && bj) {
      float sl, cl; __sincosf(la, &sl, &cl);
      gr = -cl * s; gi = -sl * s;              // -e^{il} sin(t/2)
    } else if (bi && !bj) {
      float sp, cp; __sincosf(ph, &sp, &cp);
      gr = cp * s; gi = sp * s;                // e^{ip} sin(t/2)
    } else {
      float sc, cc; __sincosf(ph + la, &sc, &cc);
      gr = cc * c; gi = sc * c;                // e^{i(p+l)} cos(t/2)
    }
    float nre = re * gr - im * gi;
    float nim = re * gi + im * gr;
    re = nre; im = nim;
  }
  float* base = ws + mat * 8192;
  base[idx] = re;          // real plane
  base[4096 + idx] = im;   // imag plane
}

#define LSTR 66  // padded LDS row stride (floats): stride-66 columns spread across banks

__global__ __launch_bounds__(128) void u3_apply(const float* __restrict__ x,
                                                const float* __restrict__ ws,
                                                float* __restrict__ out) {
  __shared__ float Xs[64 * LSTR];
  __shared__ float Tr[64 * LSTR];
  __shared__ float Ti[64 * LSTR];

  const int b    = blockIdx.x;      // batch element
  const int tid  = threadIdx.x;
  const int wave = tid >> 5;        // 0..3 (wave32)
  const int lane = tid & 31;
  const int half = lane >> 4;       // 0/1: lanes 0-15 vs 16-31
  const int l16  = lane & 15;

  const float* __restrict__ Ar = ws;
  const float* __restrict__ Ai = ws + 4096;
  const float* __restrict__ Br = ws + 8192;
  const float* __restrict__ Bi = ws + 12288;

  // Warm the WGP caches for the B planes (global_prefetch_b8) while stage 1 runs.
  __builtin_prefetch(Br + tid * 32, 0, 1);
  __builtin_prefetch(Bi + tid * 32, 0, 1);

  // ---- Stage X (64x64 real f32) into padded LDS via the async copy engine ----
  // global_load_async_to_lds_b128: GVS mode (SGPR64 base + 32-bit lane offset),
  // tracked by ASYNCcnt; no VGPR round-trip. 16B chunk stays inside one padded row.
  {
    const float* xb = x + b * 4096;            // uniform -> SGPR pair
    unsigned ldsXs = (unsigned)(size_t)(void*)Xs;
#pragma unroll
    for (int it = 0; it < 8; ++it) {
      int e4   = it * 128 + tid;               // float4 index 0..1023
      int row  = e4 >> 4;
      int col4 = e4 & 15;
      unsigned lds_addr = ldsXs + (unsigned)((row * LSTR + col4 * 4) * 4);
      unsigned goff     = (unsigned)(e4 * 16); // byte offset into xb
      asm volatile("global_load_async_to_lds_b128 %0, %1, %2"
                   :: "v"(lds_addr), "v"(goff), "s"(xb)
                   : "memory");
    }
    asm volatile("s_wait_asynccnt 0x0" ::: "memory");
  }
  __syncthreads();

  const int row0 = wave * 16;  // this wave's 16-row output strip

  // ---------------- Stage 1: T = A * X (complex = real matrix x2) ----------------
  {
    v8f accR[4] = {{0,0,0,0,0,0,0,0},{0,0,0,0,0,0,0,0},{0,0,0,0,0,0,0,0},{0,0,0,0,0,0,0,0}};
    v8f accI[4] = {{0,0,0,0,0,0,0,0},{0,0,0,0,0,0,0,0},{0,0,0,0,0,0,0,0},{0,0,0,0,0,0,0,0}};
    for (int k0 = 0; k0 < 64; k0 += 4) {
      // A fragment 16x4: lanes 0-15 -> K = k0,k0+1 ; lanes 16-31 -> K = k0+2,k0+3
      const int arow = row0 + l16;
      const int ak   = k0 + 2 * half;
      v2f aR, aI;
      aR.x = Ar[arow * 64 + ak];     aR.y = Ar[arow * 64 + ak + 1];
      aI.x = Ai[arow * 64 + ak];     aI.y = Ai[arow * 64 + ak + 1];
#pragma unroll
      for (int n = 0; n < 4; ++n) {
        const int col0 = n * 16;
        // B fragment 4x16: VGPR0 rows {k0+2h}, VGPR1 rows {k0+2h+1}, lane -> column
        v2f bx;
        bx.x = Xs[(k0 + 2 * half)     * LSTR + col0 + l16];
        bx.y = Xs[(k0 + 2 * half + 1) * LSTR + col0 + l16];
        accR[n] = __builtin_amdgcn_wmma_f32_16x16x4_f32(false, aR, false, bx,
                                                        (short)0, accR[n], false, false);
        accI[n] = __builtin_amdgcn_wmma_f32_16x16x4_f32(false, aI, false, bx,
                                                        (short)0, accI[n], false, false);
      }
    }
    // Park T in LDS. C/D layout: VGPR v -> row v (lanes 0-15) / v+8 (lanes 16-31).
#pragma unroll
    for (int n = 0; n < 4; ++n) {
      const int col0 = n * 16;
#pragma unroll
      for (int v = 0; v < 8; ++v) {
        const int r = row0 + v + 8 * half;
        Tr[r * LSTR + col0 + l16] = accR[n][v];
        Ti[r * LSTR + col0 + l16] = accI[n][v];
      }
    }
  }
  __syncthreads();

  // ---------------- Stage 2: Y = T * B^T (full complex) ----------------
  {
    v8f accR[4] = {{0,0,0,0,0,0,0,0},{0,0,0,0,0,0,0,0},{0,0,0,0,0,0,0,0},{0,0,0,0,0,0,0,0}};
    v8f accI[4] = {{0,0,0,0,0,0,0,0},{0,0,0,0,0,0,0,0},{0,0,0,0,0,0,0,0},{0,0,0,0,0,0,0,0}};
    for (int k0 = 0; k0 < 64; k0 += 4) {
      const int trow = row0 + l16;
      const int tk   = k0 + 2 * half;
      v2f tR, tI;
      tR.x = Tr[trow * LSTR + tk];   tR.y = Tr[trow * LSTR + tk + 1];
      tI.x = Ti[trow * LSTR + tk];   tI.y = Ti[trow * LSTR + tk + 1];
#pragma unroll
      for (int n = 0; n < 4; ++n) {
        const int j = n * 16 + l16;  // output column this lane contributes to
        // B-operand[k, n] = B[j, k]  (B^T): read B row j at columns tk, tk+1
        v2f bR, bI;
        bR.x = Br[j * 64 + tk];      bR.y = Br[j * 64 + tk + 1];
        bI.x = Bi[j * 64 + tk];      bI.y = Bi[j * 64 + tk + 1];
        v2f bIneg = -bI;             // f32 WMMA has no A/B NEG modifier -> negate operand
        // Yr += Tr*BrT - Ti*BiT ; Yi += Tr*BiT + Ti*BrT
        accR[n] = __builtin_amdgcn_wmma_f32_16x16x4_f32(false, tR, false, bR,
                                                        (short)0, accR[n], false, false);
        accR[n] = __builtin_amdgcn_wmma_f32_16x16x4_f32(false, tI, false, bIneg,
                                                        (short)0, accR[n], false, false);
        accI[n] = __builtin_amdgcn_wmma_f32_16x16x4_f32(false, tR, false, bI,
                                                        (short)0, accI[n], false, false);
        accI[n] = __builtin_amdgcn_wmma_f32_16x16x4_f32(false, tI, false, bR,
                                                        (short)0, accI[n], false, false);
      }
    }
    // Write complex64 output: interleaved (re, im) float pairs, contiguous float2 per lane.
    float2* ob = (float2*)out + b * 4096;
#pragma unroll
    for (int n = 0; n < 4; ++n) {
#pragma unroll
      for (int v = 0; v < 8; ++v) {
        const int r = row0 + v + 8 * half;
        const int j = n * 16 + l16;
        float2 val;
        val.x = accR[n][v];
        val.y = accI[n][v];
        ob[r * 64 + j] = val;
      }
    }
  }
}

extern "C" void kernel_launch(void* const* d_in, const int* in_sizes, int n_in,
                              void* d_out, int out_size, void* d_ws, size_t ws_size,
                              hipStream_t stream) {
  (void)in_sizes; (void)n_in; (void)out_size; (void)ws_size;
  const float* x      = (const float*)d_in[0];   // (64, 4096, 1) f32
  const float* thetas = (const float*)d_in[1];   // (12, 3) f32
  float* ws  = (float*)d_ws;                     // 16384 floats used
  float* out = (float*)d_out;                    // 64*4096 complex64 = 524288 floats

  u3_build_ops<<<32, 256, 0, stream>>>(thetas, ws);
  u3_apply<<<64, 128, 0, stream>>>(x, ws, out);
}